// HierarchicalSparseAttentionTriton_24618752541184
// MI455X (gfx1250) — compile-verified
//
#include <hip/hip_runtime.h>
#include <hip/hip_bf16.h>

typedef float v2f __attribute__((ext_vector_type(2)));
typedef float v8f __attribute__((ext_vector_type(8)));
typedef int   v4i __attribute__((ext_vector_type(4)));

// ---- CDNA5 async global->LDS path (compile-safe guard) ----
#ifndef USE_ASYNC_LDS
  #if defined(__has_builtin)
    #if __has_builtin(__builtin_amdgcn_global_load_async_to_lds_b128) && \
        __has_builtin(__builtin_amdgcn_s_wait_asynccnt)
      #define USE_ASYNC_LDS 1
    #else
      #define USE_ASYNC_LDS 0
    #endif
  #else
    #define USE_ASYNC_LDS 0
  #endif
#endif

namespace {
constexpr int Bc = 4, Nn = 4096, Hh = 16, Dd = 64;
constexpr int TILE_N = 16;
constexpr int NT    = Nn / TILE_N;       // 256 tiles along n
constexpr int ROWF  = Hh * Dd;           // 1024 floats between consecutive n rows
constexpr int LSTR  = 68;                // padded LDS row stride: banks (4M+j)%64 all distinct
constexpr int TILE_F  = 16 * LSTR;       // 1088 floats per 16x64 tile
constexpr int SCR_OFF = 4 * TILE_F;      // accumulator dump scratch (3*256 floats)
constexpr int WL_OFF  = SCR_OFF + 768;   // softmax weights broadcast (16*4 floats)
constexpr int WAVE_LDS = WL_OFF + 64;    // 5184 floats per wave
constexpr float SM_SCALE = 0.125f;       // 1/sqrt(64)
constexpr float EPSF = 1e-9f;
}

__device__ __forceinline__ void cp16(const float* g, float* l) {
#if USE_ASYNC_LDS
  __builtin_amdgcn_global_load_async_to_lds_b128(
      (__attribute__((address_space(1))) v4i*)(g),
      (__attribute__((address_space(3))) v4i*)(l), 0, 0);
#else
  *(float4*)l = *(const float4*)g;
#endif
}

__global__ void __launch_bounds__(64)
hsa_wmma_kernel(const float* __restrict__ Qp, const float* __restrict__ Kp,
                const float* __restrict__ Vp, const float* __restrict__ Kc,
                const float* __restrict__ Vc, float* __restrict__ outO,
                float* __restrict__ outW) {
  __shared__ float lds[2 * WAVE_LDS];
  const int lane = threadIdx.x & 31;
  const int wave = threadIdx.x >> 5;
  float* L = lds + wave * WAVE_LDS;

  const int gw = blockIdx.x * 2 + wave;       // one wave per (b, h, 16-node tile)
  const int nt = gw % NT;
  const int h  = (gw / NT) % Hh;
  const int b  = gw / (NT * Hh);
  const int n0 = nt * TILE_N;

  const long pbase = ((long)(b * Nn + n0) * Hh + h) * Dd;           // parent row 0
  const long cbase = ((long)(b * 2 * Nn + 2 * n0) * Hh + h) * Dd;   // child row 0

  float* qT  = L;
  float* kpT = L + TILE_F;
  float* keT = L + 2 * TILE_F;   // even children (child 0 of each node)
  float* koT = L + 3 * TILE_F;   // odd children  (child 1 of each node)
  float* scr = L + SCR_OFF;
  float* wl  = L + WL_OFF;

  // ---- stage Q / K tiles into LDS (b128 granules, coalesced / async) ----
  for (int it = 0; it < 8; ++it) {
    int idx = it * 32 + lane;                 // 256 float4 per 16x64 tile
    int row = idx >> 4;
    int c4  = (idx & 15) << 2;
    cp16(Qp + pbase + (long)row * ROWF + c4, qT + row * LSTR + c4);
    cp16(Kp + pbase + (long)row * ROWF + c4, kpT + row * LSTR + c4);
  }
  for (int it = 0; it < 16; ++it) {           // 32 child rows, deinterleave even/odd
    int idx = it * 32 + lane;
    int r   = idx >> 4;
    int c4  = (idx & 15) << 2;
    float* dst = ((r & 1) ? koT : keT) + (r >> 1) * LSTR + c4;
    cp16(Kc + cbase + (long)r * ROWF + c4, dst);
  }
#if USE_ASYNC_LDS
  __builtin_amdgcn_s_wait_asynccnt(0);
#endif

  // ---- scores via v_wmma_f32_16x16x4_f32 chains: S = Q * K^T, use diagonal ----
  // A 16x4 layout: lanes 0-15 hold M=lane K={k,k+1}; lanes 16-31 hold M=lane-16 K={k+2,k+3}
  const int m     = lane & 15;
  const int khalf = (lane >> 4) << 1;
  v8f accS = {}, accE = {}, accO = {};
  for (int c = 0; c < 16; ++c) {
    const int kb = (c << 2) + khalf;
    v2f a  = *(const v2f*)(qT  + m * LSTR + kb);
    v2f bp = *(const v2f*)(kpT + m * LSTR + kb);
    v2f be = *(const v2f*)(keT + m * LSTR + kb);
    v2f bo = *(const v2f*)(koT + m * LSTR + kb);
    accS = __builtin_amdgcn_wmma_f32_16x16x4_f32(false, a, false, bp, (short)0, accS, false, false);
    accE = __builtin_amdgcn_wmma_f32_16x16x4_f32(false, a, false, be, (short)0, accE, false, false);
    accO = __builtin_amdgcn_wmma_f32_16x16x4_f32(false, a, false, bo, (short)0, accO, false, false);
  }

  // ---- dump accumulators, pull diagonal (M=N=i), 3-way softmax ----
  for (int r = 0; r < 8; ++r) {
    scr[r * 32 + lane]       = accS[r];
    scr[256 + r * 32 + lane] = accE[r];
    scr[512 + r * 32 + lane] = accO[r];
  }
  if (lane < 16) {
    const int i   = lane;
    const int loc = (i & 7) * 32 + ((i < 8) ? i : (i + 16)); // C/D layout: VGPR i%8, lane i or i+16
    float s0 = scr[loc]       * SM_SCALE;
    float s1 = scr[256 + loc] * SM_SCALE;
    float s2 = scr[512 + loc] * SM_SCALE;
    float mx = fmaxf(s0, fmaxf(s1, s2));
    float e0 = __expf(s0 - mx), e1 = __expf(s1 - mx), e2 = __expf(s2 - mx);
    float inv = 1.0f / (e0 + e1 + e2 + EPSF);
    float w0 = e0 * inv, w1 = e1 * inv, w2 = e2 * inv;
    long wb = ((long)(b * Nn + n0 + i) * Hh + h) * 3;
    outW[wb]     = w0;
    outW[wb + 1] = w1;
    outW[wb + 2] = w2;
    wl[i * 4]     = w0;
    wl[i * 4 + 1] = w1;
    wl[i * 4 + 2] = w2;
  }

  // ---- weighted V combine: lane spans 2 contiguous d columns -> b64 coalesced ----
  const int d0 = lane << 1;
  for (int i = 0; i < 16; ++i) {
    float w0 = wl[i * 4], w1 = wl[i * 4 + 1], w2 = wl[i * 4 + 2];
    const float2 vp = *(const float2*)(Vp + pbase + (long)i * ROWF + d0);
    const float2 ve = *(const float2*)(Vc + cbase + (long)(2 * i) * ROWF + d0);
    const float2 vo = *(const float2*)(Vc + cbase + (long)(2 * i + 1) * ROWF + d0);
    float2 o;
    o.x = w0 * vp.x + w1 * ve.x + w2 * vo.x;
    o.y = w0 * vp.y + w1 * ve.y + w2 * vo.y;
    *(float2*)(outO + pbase + (long)i * ROWF + d0) = o;
  }
}

extern "C" void kernel_launch(void* const* d_in, const int* in_sizes, int n_in,
                              void* d_out, int out_size, void* d_ws, size_t ws_size,
                              hipStream_t stream) {
  (void)in_sizes; (void)n_in; (void)out_size; (void)d_ws; (void)ws_size;
  const float* Qp = (const float*)d_in[0];
  const float* Kp = (const float*)d_in[1];
  const float* Vp = (const float*)d_in[2];
  const float* Kc = (const float*)d_in[3];
  const float* Vc = (const float*)d_in[4];
  float* out = (float*)d_out;
  float* w   = out + (size_t)Bc * Nn * Hh * Dd;   // second return value, concatenated
  const int totalWaves = Bc * Hh * NT;            // 16384
  dim3 grid(totalWaves / 2), block(64);           // 2 waves per block
  hipLaunchKernelGGL(hsa_wmma_kernel, grid, block, 0, stream, Qp, Kp, Vp, Kc, Vc, out, w);
}